// PatchStatisticsLoss_2010044694931
// MI455X (gfx1250) — compile-verified
//
#include <hip/hip_runtime.h>

// ---------------------------------------------------------------------------
// PatchStatisticsLoss for MI455X (gfx1250, wave32)
//   pred/target: [8,3,1024,1024] f32, PATCH=16, STRIDE=8 -> 127x127 locations
//   per (b,c,dy,dx): mean & biased var over strided subgrid; loss = MSE(m)+MSE(v)
// ---------------------------------------------------------------------------

typedef __attribute__((ext_vector_type(2))) float v2f;
typedef __attribute__((ext_vector_type(8))) float v8f;

#define IMG 1024
#define NIMG 24          // 8*3 images per tensor
#define ROWS_TOTAL 49152 // 2*8*3*1024
#define INV_L (1.0f / 16129.0f)   // 127*127 locations
#define INV_BINS (1.0f / 6144.0f) // 8*768 (b,k) pairs

// ---------------------------------------------------------------------------
// Kernel 1: one wave32 per image row. Each pixel read exactly once.
// Produces per-row S[dx] (strided col-sum, dx=0..15) and Q[dx] (sum of squares)
// ws layout: ws[rowIdx*32 + d] : d<16 -> S[d], d>=16 -> Q[d-16]
//   rowIdx = t*24576 + img*1024 + y   (t: 0=pred 1=target, img=b*3+c)
// ---------------------------------------------------------------------------
__global__ __launch_bounds__(256) void row_stats_kernel(
    const float* __restrict__ pred, const float* __restrict__ targ,
    float* __restrict__ ws) {
  const int lane = threadIdx.x & 31;
  const int wv   = threadIdx.x >> 5;
  const int gi   = blockIdx.x * 8 + wv;          // 0..49151
  const int t    = gi / (NIMG * IMG);
  const int rem  = gi - t * (NIMG * IMG);        // img*1024 + y
  const float* __restrict__ src = t ? targ : pred;
  const float* __restrict__ rowp = src + (size_t)rem * IMG;

  // Prefetch the whole 4KB row (32 lanes x 128B spacing) -> global_prefetch_b8
  __builtin_prefetch(rowp + lane * 32, 0, 1);

  // Per-lane residue-class partials. col = 128*i + 4*lane + j,
  // col%8 = 4*(lane&1)+j : even lanes own residues 0..3, odd lanes 4..7.
  const float4* __restrict__ row4 = (const float4*)rowp;
  float as0 = 0.f, as1 = 0.f, as2 = 0.f, as3 = 0.f;
  float aq0 = 0.f, aq1 = 0.f, aq2 = 0.f, aq3 = 0.f;
#pragma unroll
  for (int i = 0; i < 8; ++i) {
    float4 v = row4[i * 32 + lane];
    as0 += v.x; aq0 = fmaf(v.x, v.x, aq0);
    as1 += v.y; aq1 = fmaf(v.y, v.y, aq1);
    as2 += v.z; aq2 = fmaf(v.z, v.z, aq2);
    as3 += v.w; aq3 = fmaf(v.w, v.w, aq3);
  }

  // Butterfly among same-parity lanes (xor offsets keep bit0): after this,
  // every even lane holds T[0..3], every odd lane T[4..7] (and Q likewise).
#pragma unroll
  for (int off = 2; off < 32; off <<= 1) {
    as0 += __shfl_xor(as0, off, 32); aq0 += __shfl_xor(aq0, off, 32);
    as1 += __shfl_xor(as1, off, 32); aq1 += __shfl_xor(aq1, off, 32);
    as2 += __shfl_xor(as2, off, 32); aq2 += __shfl_xor(aq2, off, 32);
    as3 += __shfl_xor(as3, off, 32); aq3 += __shfl_xor(aq3, off, 32);
  }

  // Broadcast the 8 class sums to all lanes (lane0: r=0..3, lane1: r=4..7)
  float t0 = __shfl(as0, 0, 32), t1 = __shfl(as1, 0, 32);
  float t2 = __shfl(as2, 0, 32), t3 = __shfl(as3, 0, 32);
  float t4 = __shfl(as0, 1, 32), t5 = __shfl(as1, 1, 32);
  float t6 = __shfl(as2, 1, 32), t7 = __shfl(as3, 1, 32);
  float u0 = __shfl(aq0, 0, 32), u1 = __shfl(aq1, 0, 32);
  float u2 = __shfl(aq2, 0, 32), u3 = __shfl(aq3, 0, 32);
  float u4 = __shfl(aq0, 1, 32), u5 = __shfl(aq1, 1, 32);
  float u6 = __shfl(aq2, 1, 32), u7 = __shfl(aq3, 1, 32);

  const int r = lane & 7;
  float Tr = (r == 0) ? t0 : (r == 1) ? t1 : (r == 2) ? t2 : (r == 3) ? t3
           : (r == 4) ? t4 : (r == 5) ? t5 : (r == 6) ? t6 : t7;
  float Qr = (r == 0) ? u0 : (r == 1) ? u1 : (r == 2) ? u2 : (r == 3) ? u3
           : (r == 4) ? u4 : (r == 5) ? u5 : (r == 6) ? u6 : u7;

  // Endpoint correction: dx=r excludes col r+1016; dx=r+8 excludes col r.
  const int dxlo = (lane & 8) == 0;               // (lane&15) < 8
  float e = rowp[dxlo ? (r + 1016) : r];
  float val = (lane < 16) ? (Tr - e) : (Qr - e * e);
  ws[(size_t)gi * 32 + lane] = val;               // one coalesced 128B store
}

// ---------------------------------------------------------------------------
// Kernel 2: one block per image pair (b,c); wave q = row-residue class q.
// U[q][dx] = sum over 128 class rows of S -> ones(16x4) x S-chunk(4x16) WMMA,
// accumulated over 32 chunks in V_WMMA_F32_16X16X4_F32 (exact f32 sums;
// result is the column sum in every D row, robust to K-slot mapping).
// Then bins: dy=q   -> U - S[y=q+1016];  dy=q+8 -> U - S[y=q].
// ---------------------------------------------------------------------------
__global__ __launch_bounds__(256) void bin_reduce_kernel(
    const float* __restrict__ ws, float* __restrict__ out) {
  __shared__ float wsum[8];
  const int lane = threadIdx.x & 31;
  const int q    = threadIdx.x >> 5;   // class 0..7
  const int img  = blockIdx.x;         // 0..23

  const float* __restrict__ wp = ws + (size_t)img * IMG * 32;            // pred
  const float* __restrict__ wt = ws + (size_t)(NIMG * IMG + img * IMG) * 32; // target

  const int n   = lane & 15;       // matrix column (dx)
  const int h   = lane >> 4;       // lane half -> which K slots
  const int kk0 = 2 * h;           // rows of each 4-row chunk held by this lane
  const int kk1 = 2 * h + 1;

  v2f a; a[0] = 1.0f; a[1] = 1.0f; // ones A-matrix (16x4)
  v8f cps = {}; v8f cpq = {}; v8f cts = {}; v8f ctq = {};

  for (int ch = 0; ch < 32; ++ch) {
    const int y0 = 8 * (4 * ch + kk0) + q;
    const int y1 = 8 * (4 * ch + kk1) + q;
    const int o0 = y0 * 32 + n, o1 = y1 * 32 + n;
    v2f bps, bpq, bts, btq;
    bps[0] = wp[o0];      bps[1] = wp[o1];
    bpq[0] = wp[o0 + 16]; bpq[1] = wp[o1 + 16];
    bts[0] = wt[o0];      bts[1] = wt[o1];
    btq[0] = wt[o0 + 16]; btq[1] = wt[o1 + 16];
    cps = __builtin_amdgcn_wmma_f32_16x16x4_f32(false, a, false, bps, (short)0, cps, false, false);
    cpq = __builtin_amdgcn_wmma_f32_16x16x4_f32(false, a, false, bpq, (short)0, cpq, false, false);
    cts = __builtin_amdgcn_wmma_f32_16x16x4_f32(false, a, false, bts, (short)0, cts, false, false);
    ctq = __builtin_amdgcn_wmma_f32_16x16x4_f32(false, a, false, btq, (short)0, ctq, false, false);
  }

  // Every lane's c[0] = column sum for column n = lane&15.
  const float Usp = cps[0], Uqp = cpq[0], Ust = cts[0], Uqt = ctq[0];

  float partial = 0.0f;
  if (lane < 16) {
    const int d = lane;
#pragma unroll
    for (int aa = 0; aa < 2; ++aa) {
      const int yex = aa ? q : (q + 1016);   // aa=0 -> dy=q ; aa=1 -> dy=q+8
      const int oe  = yex * 32 + d;
      float mp = (Usp - wp[oe]) * INV_L;
      float vp = (Uqp - wp[oe + 16]) * INV_L - mp * mp;
      float mt = (Ust - wt[oe]) * INV_L;
      float vt = (Uqt - wt[oe + 16]) * INV_L - mt * mt;
      float dm = mp - mt, dv = vp - vt;
      partial += dm * dm + dv * dv;
    }
  }
#pragma unroll
  for (int off = 1; off < 32; off <<= 1)
    partial += __shfl_xor(partial, off, 32);
  if (lane == 0) wsum[q] = partial;
  __syncthreads();
  if (threadIdx.x == 0) {
    float total = 0.0f;
#pragma unroll
    for (int i = 0; i < 8; ++i) total += wsum[i];
    atomicAdd(out, total * INV_BINS);
  }
}

__global__ void zero_out_kernel(float* out) {
  if (threadIdx.x == 0) out[0] = 0.0f;
}

extern "C" void kernel_launch(void* const* d_in, const int* in_sizes, int n_in,
                              void* d_out, int out_size, void* d_ws, size_t ws_size,
                              hipStream_t stream) {
  const float* pred = (const float*)d_in[0];
  const float* targ = (const float*)d_in[1];
  float* out = (float*)d_out;
  float* ws  = (float*)d_ws;   // needs 49152*32*4 = 6.29 MB

  zero_out_kernel<<<1, 32, 0, stream>>>(out);
  row_stats_kernel<<<ROWS_TOTAL / 8, 256, 0, stream>>>(pred, targ, ws);
  bin_reduce_kernel<<<NIMG, 256, 0, stream>>>(ws, out);
}